// FraudQMLModel_65481071400723
// MI455X (gfx1250) — compile-verified
//
#include <hip/hip_runtime.h>

typedef float v2f __attribute__((ext_vector_type(2)));
typedef float v8f __attribute__((ext_vector_type(8)));

// ---------------- complex helpers (setup kernel only, runs once) -------------
struct Cplx { float r, i; };
__device__ __forceinline__ Cplx cmul(Cplx a, Cplx b) { return Cplx{a.r*b.r - a.i*b.i, a.r*b.i + a.i*b.r}; }
__device__ __forceinline__ Cplx cadd(Cplx a, Cplx b) { return Cplx{a.r + b.r, a.i + b.i}; }
__device__ __forceinline__ Cplx cscale(float s, Cplx a) { return Cplx{s*a.r, s*a.i}; }
__device__ __forceinline__ Cplx conjc(Cplx a) { return Cplx{a.r, -a.i}; }
__device__ __forceinline__ Cplx cexpi(float t) { return Cplx{cosf(t), sinf(t)}; }

__device__ void mm4(const Cplx A[4][4], const Cplx B[4][4], Cplx O[4][4]) {
  for (int r = 0; r < 4; ++r)
    for (int c = 0; c < 4; ++c) {
      Cplx acc = {0.f, 0.f};
      for (int k = 0; k < 4; ++k) acc = cadd(acc, cmul(A[r][k], B[k][c]));
      O[r][c] = acc;
    }
}

// m = RY(tr) @ RZ(tz) = [[c*e^{-i tz/2}, -s*e^{+i tz/2}],[s*e^{-i tz/2}, c*e^{+i tz/2}]]
__device__ void ry_rz(float tr, float tz, Cplx m[2][2]) {
  float c = cosf(0.5f * tr), s = sinf(0.5f * tr);
  Cplx em = cexpi(-0.5f * tz), ep = cexpi(0.5f * tz);
  m[0][0] = cscale(c, em);  m[0][1] = cscale(-s, ep);
  m[1][0] = cscale(s, em);  m[1][1] = cscale(c, ep);
}

// Builds U = L_{n-1} @ ... @ L_0, then S = W00 * Re(diag(ph)^H (U^H Z U) diag(ph)).
// ws[0..15] = S (row major), ws[16] = b0.
__global__ void fraud_setup(const float* __restrict__ params, int n_layers,
                            const float* __restrict__ W, const float* __restrict__ bvec,
                            float* __restrict__ ws) {
  if (threadIdx.x != 0 || blockIdx.x != 0) return;
  Cplx U[4][4];
  for (int r = 0; r < 4; ++r)
    for (int c = 0; c < 4; ++c) U[r][c] = Cplx{(r == c) ? 1.f : 0.f, 0.f};

  for (int l = 0; l < n_layers; ++l) {
    float q[14];
    for (int k = 0; k < 14; ++k) {
      float lim = (k < 12) ? 5.f : 1.f;
      float v = params[l * 14 + k];
      q[k] = fminf(fmaxf(v, -lim), lim);
    }
    // T = HH @ CNOT   (HH = 0.5*sign pattern; CNOT swaps columns 2<->3)
    const float HH[4][4] = {{1,1,1,1},{1,-1,1,-1},{1,1,-1,-1},{1,-1,-1,1}};
    const int km[4] = {0, 1, 3, 2};
    Cplx T[4][4];
    for (int r = 0; r < 4; ++r)
      for (int c = 0; c < 4; ++c) T[r][c] = Cplx{0.5f * HH[r][km[c]], 0.f};
    // Diagonal P (kron of phase gates) and Smat (kron of RZ) as a combined row scale
    Cplx dP[4] = { Cplx{1.f, 0.f}, cexpi(q[3]), cexpi(q[2]), cexpi(q[2] + q[3]) };
    float t0 = q[4] + q[6], t1 = q[5] + q[7];
    Cplx dS[4] = { cexpi(-0.5f*(t0 + t1)), cexpi(-0.5f*(t0 - t1)),
                   cexpi( 0.5f*(t0 - t1)), cexpi( 0.5f*(t0 + t1)) };
    for (int r = 0; r < 4; ++r) {
      Cplx f = cmul(dP[r], dS[r]);
      for (int c = 0; c < 4; ++c) T[r][c] = cmul(f, T[r][c]);
    }
    // D = kron(RY(q10)RZ(q8), RY(q11)RZ(q9))
    Cplx m0[2][2], m1[2][2];
    ry_rz(q[10], q[8], m0);
    ry_rz(q[11], q[9], m1);
    Cplx D[4][4];
    for (int r0 = 0; r0 < 2; ++r0)
      for (int r1 = 0; r1 < 2; ++r1)
        for (int c0 = 0; c0 < 2; ++c0)
          for (int c1 = 0; c1 < 2; ++c1)
            D[2*r0 + r1][2*c0 + c1] = cmul(m0[r0][c0], m1[r1][c1]);
    Cplx T2[4][4];
    mm4(D, T, T2);
    // CP: row 3 scaled by e^{i(q12+q13)}
    Cplx cp = cexpi(q[12] + q[13]);
    for (int c = 0; c < 4; ++c) T2[3][c] = cmul(cp, T2[3][c]);
    // U = layer @ U
    Cplx Un[4][4];
    mm4(T2, U, Un);
    for (int r = 0; r < 4; ++r)
      for (int c = 0; c < 4; ++c) U[r][c] = Un[r][c];
  }

  const float z[4] = {1.f, 1.f, -1.f, -1.f};
  const Cplx ph[4] = { Cplx{1,0}, Cplx{0,-1}, Cplx{0,-1}, Cplx{-1,0} };
  float W00 = W[0];
  for (int j = 0; j < 4; ++j)
    for (int k = 0; k < 4; ++k) {
      Cplx a = {0.f, 0.f};
      for (int i = 0; i < 4; ++i)
        a = cadd(a, cscale(z[i], cmul(conjc(U[i][j]), U[i][k])));
      Cplx b = cmul(cmul(conjc(ph[j]), ph[k]), a);
      ws[4*j + k] = b.r * W00;
    }
  ws[16] = bvec[0];
}

// ---------------- streaming main kernel (WMMA quadratic form) ----------------
__device__ __forceinline__ float lane_gather(float v, int src) {
  return __int_as_float(__builtin_amdgcn_ds_bpermute(src << 2, __float_as_int(v)));
}

__global__ __launch_bounds__(256) void fraud_main(const float* __restrict__ x,
                                                  const float* __restrict__ ws,
                                                  float* __restrict__ out, int B) {
  const int lane = threadIdx.x & 31;
  const int nthreads = gridDim.x * blockDim.x;
  const int wbase0 = blockIdx.x * blockDim.x + (threadIdx.x - lane);
  const bool lo = lane < 16;

  // A operand: S embedded in the 16x4 f32 A layout.
  // VGPR0: lanes 0-15 -> (M=lane, K=0); lanes 16-31 -> (M=lane-16, K=2). VGPR1: K=1 / K=3.
  int row = lane & 15;
  int cb = (lane >> 4) << 1;
  v2f Aop;
  Aop.x = (row < 4) ? ws[4*row + cb]     : 0.0f;
  Aop.y = (row < 4) ? ws[4*row + cb + 1] : 0.0f;
  float bias = ws[16];
  v8f zeroC = {0.f, 0.f, 0.f, 0.f, 0.f, 0.f, 0.f, 0.f};

  for (int base = wbase0; base < B; base += nthreads) {   // wave-uniform loop bound
    int s = base + lane;
    if (base + 32 <= B) {                                 // full wave: EXEC all ones
      __builtin_prefetch(&x[2 * (s + 4 * nthreads)], 0, 1);
      float2 xv = reinterpret_cast<const float2*>(x)[s];
      float s0, c0, s1, c1;
      __sincosf(0.5f * xv.x, &s0, &c0);
      __sincosf(0.5f * xv.y, &s1, &c1);
      float v0 = c0 * c1, v1 = c0 * s1, v2 = s0 * c1, v3 = s0 * s1;

      // B operand (4x16): VGPR0 lanes0-15=K0 row, lanes16-31=K2 row; VGPR1 = K1/K3.
      float u2 = lane_gather(v2, lane & 15);   // v2 of sample (lane-16) for upper lanes
      float u3 = lane_gather(v3, lane & 15);
      float u0 = lane_gather(v0, lane | 16);   // v0 of sample (lane+16) for lower lanes
      float u1 = lane_gather(v1, lane | 16);
      v2f B0, B1;
      B0.x = lo ? v0 : u2;  B0.y = lo ? v1 : u3;   // samples base+0 .. base+15
      B1.x = lo ? u0 : v2;  B1.y = lo ? u1 : v3;   // samples base+16 .. base+31

      v8f d0 = __builtin_amdgcn_wmma_f32_16x16x4_f32(false, Aop, false, B0,
                                                     (short)0, zeroC, false, false);
      v8f d1 = __builtin_amdgcn_wmma_f32_16x16x4_f32(false, Aop, false, B1,
                                                     (short)0, zeroC, false, false);

      // D layout: VGPR r, lane n -> D[r][n] = (S v_n)_r. Lower lanes read d0 in place,
      // upper lanes fetch their column of d1 from lane (lane-16).
      float w0 = lane_gather(d1[0], lane & 15);
      float w1 = lane_gather(d1[1], lane & 15);
      float w2 = lane_gather(d1[2], lane & 15);
      float w3 = lane_gather(d1[3], lane & 15);
      float t0 = lo ? d0[0] : w0;
      float t1 = lo ? d0[1] : w1;
      float t2 = lo ? d0[2] : w2;
      float t3 = lo ? d0[3] : w3;

      out[s] = t0*v0 + t1*v1 + t2*v2 + t3*v3 + bias;   // logit = v^T S v (W00 folded in)
    } else if (s < B) {
      // tail: plain VALU quadratic form
      float2 xv = reinterpret_cast<const float2*>(x)[s];
      float s0, c0, s1, c1;
      __sincosf(0.5f * xv.x, &s0, &c0);
      __sincosf(0.5f * xv.y, &s1, &c1);
      float vv[4] = {c0*c1, c0*s1, s0*c1, s0*s1};
      float acc = bias;
      for (int j = 0; j < 4; ++j)
        for (int k = 0; k < 4; ++k)
          acc += ws[4*j + k] * vv[j] * vv[k];
      out[s] = acc;
    }
  }
}

extern "C" void kernel_launch(void* const* d_in, const int* in_sizes, int n_in,
                              void* d_out, int out_size, void* d_ws, size_t ws_size,
                              hipStream_t stream) {
  const float* x      = (const float*)d_in[0];
  const float* params = (const float*)d_in[1];
  const float* W      = (const float*)d_in[2];
  const float* bvec   = (const float*)d_in[3];
  float* out = (float*)d_out;
  float* ws  = (float*)d_ws;

  int B = in_sizes[0] / 2;
  int n_layers = in_sizes[1] / 14;

  fraud_setup<<<1, 32, 0, stream>>>(params, n_layers, W, bvec, ws);

  int block = 256;
  int grid = (B + block - 1) / block;
  if (grid > 4096) grid = 4096;
  if (grid < 1) grid = 1;
  fraud_main<<<grid, block, 0, stream>>>(x, ws, out, B);
}